// HeteroEdgeEncoder_70712341562148
// MI455X (gfx1250) — compile-verified
//
#include <hip/hip_runtime.h>

// ---------------------------------------------------------------------------
// HeteroEdgeEncoder for MI455X (gfx1250, wave32)
//   out[t, e, :] = MLP_enc(t)( concat(x_src[ei_t[0,e]], x_dst[ei_t[1,e]]) )
//   MLP: Linear(128->64)+ReLU, 2x Linear(64->64)+ReLU, Linear(64->64)
// All matmuls via V_WMMA_F32_16X16X4_F32 (native fp32 matrix op).
// ---------------------------------------------------------------------------

typedef float v2f __attribute__((ext_vector_type(2)));
typedef float v8f __attribute__((ext_vector_type(8)));

#define HIDDEN 64
#define TILE_E 16
#define WAVES 8
#define TILES_PER_WAVE 4
#define EDGES_PER_BLOCK (WAVES * TILES_PER_WAVE * TILE_E)  // 512

// LDS layout (float indices). Strides padded so that row-stride mod 64 == 4
// dwords -> WMMA fragment ds_load_b64: lanes 0-15 banks {4L,4L+1},
// lanes 16-31 banks {4L+2,4L+3} => conflict-free.
#define SW0 132   // W0^T [64 rows n][132] (K=128 + pad 4)
#define SWR 68    // Wr^T [64 rows n][68]  (K=64  + pad 4)
#define SX  132   // per-wave x/h tile row stride (16 rows)

#define OFF_WT0 0
#define OFF_WT1 (OFF_WT0 + 64 * SW0)
#define OFF_WT2 (OFF_WT1 + 64 * SWR)
#define OFF_WT3 (OFF_WT2 + 64 * SWR)
#define OFF_B   (OFF_WT3 + 64 * SWR)      // 4 * 64 biases
#define OFF_X   (OFF_B + 4 * 64)          // WAVES * 16 * SX
#define SMEM_FLOATS (OFF_X + WAVES * TILE_E * SX)   // 38656 floats = 151 KB

// One 16x64 layer on one wave's 16-edge tile.
//   A    : LDS activations [16][K], row stride SA
//   Wt   : LDS transposed weights [64][K], row stride SWT  (Wt[n][k] = W[k][n])
//   bias : LDS [64]
//   H    : LDS output [16][64], row stride SX
// Fragment mapping per CDNA5 ISA 7.12.2 (32-bit A 16x4 / B 4x16 / C 16x16).
template <int K, int SA, int SWT, bool RELU>
__device__ __forceinline__ void mlp_layer(const float* A, const float* Wt,
                                          const float* bias, float* H,
                                          int lane) {
  const int l15 = lane & 15;
  const int kh  = (lane >> 4) << 1;   // 0 for lanes 0-15, 2 for lanes 16-31
  v8f c[4];
  c[0] = v8f{}; c[1] = v8f{}; c[2] = v8f{}; c[3] = v8f{};

  const float* ap = A + l15 * SA + kh;
  const float* wp0 = Wt + (0 * 16 + l15) * SWT + kh;
  const float* wp1 = Wt + (1 * 16 + l15) * SWT + kh;
  const float* wp2 = Wt + (2 * 16 + l15) * SWT + kh;
  const float* wp3 = Wt + (3 * 16 + l15) * SWT + kh;

#pragma unroll
  for (int k = 0; k < K; k += 4) {
    v2f a  = *(const v2f*)(ap + k);     // A frag: K = k+kh, k+kh+1
    v2f b0 = *(const v2f*)(wp0 + k);    // B frag for N-tile 0
    v2f b1 = *(const v2f*)(wp1 + k);
    v2f b2 = *(const v2f*)(wp2 + k);
    v2f b3 = *(const v2f*)(wp3 + k);
    c[0] = __builtin_amdgcn_wmma_f32_16x16x4_f32(false, a, false, b0, (short)0, c[0], false, false);
    c[1] = __builtin_amdgcn_wmma_f32_16x16x4_f32(false, a, false, b1, (short)0, c[1], false, false);
    c[2] = __builtin_amdgcn_wmma_f32_16x16x4_f32(false, a, false, b2, (short)0, c[2], false, false);
    c[3] = __builtin_amdgcn_wmma_f32_16x16x4_f32(false, a, false, b3, (short)0, c[3], false, false);
  }

  const int mhi = (lane >> 4) << 3;   // C layout: M = i + 8*(lane>=16)
#pragma unroll
  for (int j = 0; j < 4; ++j) {
    float bb = bias[j * 16 + l15];    // N = j*16 + (lane&15), broadcast read
#pragma unroll
    for (int i = 0; i < 8; ++i) {
      float v = c[j][i] + bb;
      if (RELU) v = fmaxf(v, 0.0f);
      H[(i + mhi) * SX + j * 16 + l15] = v;   // conflict-free row-striped write
    }
  }
}

__global__ __launch_bounds__(256)
void hetero_edge_mlp(const float* __restrict__ xb, const float* __restrict__ xe,
                     const int* __restrict__ ei_bb, const int* __restrict__ ei_be,
                     const int* __restrict__ ei_eb, const int* __restrict__ ei_ee,
                     const float* __restrict__ Wf, const float* __restrict__ bfirst,
                     const float* __restrict__ Wr, const float* __restrict__ brest,
                     float* __restrict__ dout, int E) {
  __shared__ float smem[SMEM_FLOATS];

  // ---- edge-type dispatch: 0=bb(enc0) 1=be(enc1) 2=eb(enc1) 3=ee(enc2) ----
  const float* xs; const float* xd; const int* ei; int enc;
  switch (blockIdx.y) {
    case 0:  xs = xb; xd = xb; ei = ei_bb; enc = 0; break;
    case 1:  xs = xb; xd = xe; ei = ei_be; enc = 1; break;
    case 2:  xs = xe; xd = xb; ei = ei_eb; enc = 1; break;
    default: xs = xe; xd = xe; ei = ei_ee; enc = 2; break;
  }
  const int* ei_src = ei;
  const int* ei_dst = ei + E;

  // ---- cooperative load of this encoder's weights (transposed) + biases ----
  const int tid = threadIdx.x;
  {
    const float* w0 = Wf + (size_t)enc * (2 * HIDDEN) * HIDDEN;  // [128][64]
    for (int idx = tid; idx < 128 * 64; idx += 256) {
      int k = idx >> 6, n = idx & 63;
      smem[OFF_WT0 + n * SW0 + k] = w0[idx];
    }
#pragma unroll
    for (int l = 0; l < 3; ++l) {
      const float* wl = Wr + ((size_t)enc * 3 + l) * HIDDEN * HIDDEN;  // [64][64]
      float* dst = smem + OFF_WT1 + l * (64 * SWR);
      for (int idx = tid; idx < 64 * 64; idx += 256) {
        int k = idx >> 6, n = idx & 63;
        dst[n * SWR + k] = wl[idx];
      }
    }
    if (tid < 64) {
      smem[OFF_B + tid] = bfirst[enc * 64 + tid];
#pragma unroll
      for (int l = 0; l < 3; ++l)
        smem[OFF_B + 64 + l * 64 + tid] = brest[(enc * 3 + l) * 64 + tid];
    }
  }
  __syncthreads();   // weights ready; everything below is wave-private

  const int wave = tid >> 5;
  const int lane = tid & 31;
  float* xin = smem + OFF_X + wave * (TILE_E * SX);   // [16][SX]

  for (int t = 0; t < TILES_PER_WAVE; ++t) {
    int edge0 = blockIdx.x * EDGES_PER_BLOCK + (wave * TILES_PER_WAVE + t) * TILE_E;
    if (edge0 >= E) break;                 // uniform per wave
    int nrem = E - edge0; if (nrem > TILE_E) nrem = TILE_E;

    // ---- gather: x_in[e][0:64]=x_src[s], x_in[e][64:128]=x_dst[d] ----
    int myIdx = 0;                         // lanes 0-15: src idx e=lane; 16-31: dst idx e=lane-16
    if (lane < 16) {
      if (edge0 + lane < E) myIdx = ei_src[edge0 + lane];
    } else {
      if (edge0 + (lane - 16) < E) myIdx = ei_dst[edge0 + (lane - 16)];
    }
    for (int e = 0; e < nrem; ++e) {
      int s = __builtin_amdgcn_readlane(myIdx, e);
      int d = __builtin_amdgcn_readlane(myIdx, e + 16);
      // 32 lanes x float2 = one coalesced 256B row read per node
      v2f vs = *(const v2f*)(xs + (size_t)s * HIDDEN + 2 * lane);
      v2f vd = *(const v2f*)(xd + (size_t)d * HIDDEN + 2 * lane);
      *(v2f*)(xin + e * SX + 2 * lane) = vs;
      *(v2f*)(xin + e * SX + 64 + 2 * lane) = vd;
    }

    // ---- 4 layers, ping-pong between column halves of the tile ----
    // (LDS ops within a wave are in-order -> reads drain before overwrite)
    mlp_layer<128, SX, SW0, true >(xin,      smem + OFF_WT0, smem + OFF_B,       xin,      lane);
    mlp_layer< 64, SX, SWR, true >(xin,      smem + OFF_WT1, smem + OFF_B + 64,  xin + 64, lane);
    mlp_layer< 64, SX, SWR, true >(xin + 64, smem + OFF_WT2, smem + OFF_B + 128, xin,      lane);
    mlp_layer< 64, SX, SWR, false>(xin,      smem + OFF_WT3, smem + OFF_B + 192, xin + 64, lane);

    // ---- coalesced store: 8 x global_store_b128 per tile (512B each) ----
    float* outp = dout + ((size_t)blockIdx.y * (size_t)E + (size_t)edge0) * HIDDEN;
#pragma unroll
    for (int it = 0; it < 8; ++it) {
      int idx = it * 32 + lane;            // 256 float4 slots = 16 edges x 64 floats
      int e  = idx >> 4;
      int c4 = idx & 15;
      if (e < nrem) {
        float4 val = *(const float4*)(xin + e * SX + 64 + c4 * 4);
        *(float4*)(outp + (size_t)e * HIDDEN + c4 * 4) = val;
      }
    }
  }
}

extern "C" void kernel_launch(void* const* d_in, const int* in_sizes, int n_in,
                              void* d_out, int out_size, void* d_ws, size_t ws_size,
                              hipStream_t stream) {
  const float* xb    = (const float*)d_in[0];
  const float* xe    = (const float*)d_in[1];
  const int*   ei_bb = (const int*)d_in[2];
  const int*   ei_be = (const int*)d_in[3];
  const int*   ei_eb = (const int*)d_in[4];
  const int*   ei_ee = (const int*)d_in[5];
  const float* Wf    = (const float*)d_in[6];
  const float* bf    = (const float*)d_in[7];
  const float* Wr    = (const float*)d_in[8];
  const float* br    = (const float*)d_in[9];
  float* out = (float*)d_out;

  int E = in_sizes[2] / 2;   // ei_* are (2, N_EDGES)

  dim3 grid((E + EDGES_PER_BLOCK - 1) / EDGES_PER_BLOCK, 4);
  hetero_edge_mlp<<<grid, dim3(256), 0, stream>>>(
      xb, xe, ei_bb, ei_be, ei_eb, ei_ee, Wf, bf, Wr, br, out, E);
}